// GCN_33054068310762
// MI455X (gfx1250) — compile-verified
//
#include <hip/hip_runtime.h>
#include <math.h>

typedef float v2f __attribute__((ext_vector_type(2)));
typedef float v8f __attribute__((ext_vector_type(8)));

static __device__ __forceinline__ float eluf(float v) {
    return v > 0.0f ? v : (expf(v) - 1.0f);
}

// ---------------- setup kernels ----------------

__global__ void k_zero(int* __restrict__ degi, float* __restrict__ psum,
                       float* __restrict__ pcnt, int n, int G) {
    int i = blockIdx.x * blockDim.x + threadIdx.x;
    if (i < n) degi[i] = 0;
    if (i < G * 4) psum[i] = 0.0f;
    if (i < G) pcnt[i] = 0.0f;
}

__global__ void k_deg(const int* __restrict__ col, int* __restrict__ degi, int e) {
    int i = blockIdx.x * blockDim.x + threadIdx.x;
    if (i < e) atomicAdd(&degi[col[i]], 1);
}

__global__ void k_dinv(const int* __restrict__ degi, float* __restrict__ dinv, int n) {
    int i = blockIdx.x * blockDim.x + threadIdx.x;
    if (i < n) dinv[i] = rsqrtf((float)(degi[i] + 1));  // +1 = self loop, always >0
}

// ---------------- GEMM1: t1 = x[ n x 128 ] @ W1[ 128 x 64 ], f32 WMMA 16x16x4 ----------------
// Per block: 8 waves, each wave computes a 16x64 tile (4 N-tiles of 16, K loop of 32 steps).
// W1 staged in LDS column-major with stride 130 (floats) for conflict-free b64 reads.

__global__ __launch_bounds__(256) void k_gemm1(const float* __restrict__ x,
                                               const float* __restrict__ W1,
                                               float* __restrict__ t1, int n) {
    __shared__ float Wl[64 * 130];  // Wl[c*130 + k] = W1[k*64 + c]
    for (int i = threadIdx.x; i < 128 * 64; i += 256) {
        int k = i >> 6, c = i & 63;
        Wl[c * 130 + k] = W1[i];
    }
    __syncthreads();

    const int wave = threadIdx.x >> 5;
    const int lane = threadIdx.x & 31;
    const int hi   = lane >> 4;   // 0: K,K+1   1: K+2,K+3 (ISA 16x4 f32 A layout)
    const int lo   = lane & 15;   // M for A, N for B/C/D
    const long rowBase = (long)blockIdx.x * 128 + (long)wave * 16;

    long arow = rowBase + lo;
    if (arow >= n) arow = n - 1;          // clamp: keep EXEC all-1s through WMMA
    const float* xr = x + arow * 128;

    v8f acc0 = {}, acc1 = {}, acc2 = {}, acc3 = {};
    for (int k = 0; k < 128; k += 4) {
        const int ka = k + 2 * hi;
        v2f a  = *(const v2f*)(xr + ka);
        v2f b0 = *(const v2f*)(&Wl[(lo     ) * 130 + ka]);
        v2f b1 = *(const v2f*)(&Wl[(lo + 16) * 130 + ka]);
        v2f b2 = *(const v2f*)(&Wl[(lo + 32) * 130 + ka]);
        v2f b3 = *(const v2f*)(&Wl[(lo + 48) * 130 + ka]);
        acc0 = __builtin_amdgcn_wmma_f32_16x16x4_f32(false, a, false, b0, (short)0, acc0, false, false);
        acc1 = __builtin_amdgcn_wmma_f32_16x16x4_f32(false, a, false, b1, (short)0, acc1, false, false);
        acc2 = __builtin_amdgcn_wmma_f32_16x16x4_f32(false, a, false, b2, (short)0, acc2, false, false);
        acc3 = __builtin_amdgcn_wmma_f32_16x16x4_f32(false, a, false, b3, (short)0, acc3, false, false);
    }

#pragma unroll
    for (int j = 0; j < 8; ++j) {          // D layout: VGPR j -> row j + 8*hi, col lo
        long r = rowBase + j + 8 * hi;
        if (r < n) {
            float* o = t1 + r * 64 + lo;
            o[0]  = acc0[j];
            o[16] = acc1[j];
            o[32] = acc2[j];
            o[48] = acc3[j];
        }
    }
}

// ---------------- GEMM2: t2 = h1[ n x 64 ] @ W2[ 64 x 4 ] (N padded to 16) ----------------

__global__ __launch_bounds__(256) void k_gemm2(const float* __restrict__ h1,
                                               const float* __restrict__ W2,
                                               float* __restrict__ t2, int n) {
    __shared__ float Wl[16 * 66];   // Wl[c*66 + k]; cols 4..15 zero
    for (int i = threadIdx.x; i < 16 * 66; i += 256) Wl[i] = 0.0f;
    __syncthreads();
    for (int i = threadIdx.x; i < 64 * 4; i += 256) {
        int k = i >> 2, c = i & 3;
        Wl[c * 66 + k] = W2[i];
    }
    __syncthreads();

    const int wave = threadIdx.x >> 5;
    const int lane = threadIdx.x & 31;
    const int hi   = lane >> 4;
    const int lo   = lane & 15;
    const long rowBase = (long)blockIdx.x * 128 + (long)wave * 16;

    long arow = rowBase + lo;
    if (arow >= n) arow = n - 1;
    const float* hr = h1 + arow * 64;

    v8f acc = {};
    for (int k = 0; k < 64; k += 4) {
        const int ka = k + 2 * hi;
        v2f a = *(const v2f*)(hr + ka);
        v2f b = *(const v2f*)(&Wl[lo * 66 + ka]);
        acc = __builtin_amdgcn_wmma_f32_16x16x4_f32(false, a, false, b, (short)0, acc, false, false);
    }

#pragma unroll
    for (int j = 0; j < 8; ++j) {
        long r = rowBase + j + 8 * hi;
        if (r < n && lo < 4) t2[r * 4 + lo] = acc[j];
    }
}

// ---------------- self-loop init: agg = t * dinv[node]^2 ----------------

__global__ void k_init64(const float* __restrict__ t1, const float* __restrict__ dinv,
                         float* __restrict__ agg1, int n) {
    long i = (long)blockIdx.x * blockDim.x + threadIdx.x;   // n*16 float4 chunks
    if (i >= (long)n * 16) return;
    int node = (int)(i >> 4);
    float d = dinv[node];
    float s = d * d;
    float4 v = ((const float4*)t1)[i];
    float4 o; o.x = v.x * s; o.y = v.y * s; o.z = v.z * s; o.w = v.w * s;
    ((float4*)agg1)[i] = o;
}

__global__ void k_init4(const float* __restrict__ t2, const float* __restrict__ dinv,
                        float* __restrict__ agg2, int n) {
    int i = blockIdx.x * blockDim.x + threadIdx.x;
    if (i >= n) return;
    float d = dinv[i];
    float s = d * d;
    float4 v = ((const float4*)t2)[i];
    float4 o; o.x = v.x * s; o.y = v.y * s; o.z = v.z * s; o.w = v.w * s;
    ((float4*)agg2)[i] = o;
}

// ---------------- edge scatter: agg[col] += t[row] * dinv[row]*dinv[col] ----------------
// Layer 1: 16 threads per edge, float4 each -> coalesced 64-float row move.

__global__ void k_scatter1(const int* __restrict__ row, const int* __restrict__ col,
                           const float* __restrict__ dinv, const float* __restrict__ t1,
                           float* __restrict__ agg1, int e) {
    long idx = (long)blockIdx.x * blockDim.x + threadIdx.x;
    long ei = idx >> 4;
    if (ei >= e) return;
    int part = (int)(idx & 15);
    int r = row[ei], c = col[ei];
    float nrm = dinv[r] * dinv[c];
    float4 v = *(const float4*)(t1 + (long)r * 64 + part * 4);
    float* o = agg1 + (long)c * 64 + part * 4;
    unsafeAtomicAdd(o + 0, v.x * nrm);
    unsafeAtomicAdd(o + 1, v.y * nrm);
    unsafeAtomicAdd(o + 2, v.z * nrm);
    unsafeAtomicAdd(o + 3, v.w * nrm);
}

__global__ void k_scatter2(const int* __restrict__ row, const int* __restrict__ col,
                           const float* __restrict__ dinv, const float* __restrict__ t2,
                           float* __restrict__ agg2, int e) {
    long ei = (long)blockIdx.x * blockDim.x + threadIdx.x;
    if (ei >= e) return;
    int r = row[ei], c = col[ei];
    float nrm = dinv[r] * dinv[c];
    float4 v = *(const float4*)(t2 + (long)r * 4);
    float* o = agg2 + (long)c * 4;
    unsafeAtomicAdd(o + 0, v.x * nrm);
    unsafeAtomicAdd(o + 1, v.y * nrm);
    unsafeAtomicAdd(o + 2, v.z * nrm);
    unsafeAtomicAdd(o + 3, v.w * nrm);
}

// ---------------- elementwise: h1 = elu(agg1 + b1) in place ----------------

__global__ void k_elu1(float* __restrict__ agg1, const float* __restrict__ b1, long total) {
    long i = (long)blockIdx.x * blockDim.x + threadIdx.x;
    if (i >= total) return;
    int c = (int)(i & 63);
    agg1[i] = eluf(agg1[i] + b1[c]);
}

// ---------------- pooling: per node elu(agg2+b2), atomic into graph sums ----------------

__global__ void k_pool(const float* __restrict__ agg2, const float* __restrict__ b2,
                       const int* __restrict__ batch, float* __restrict__ psum,
                       float* __restrict__ pcnt, int n) {
    int i = blockIdx.x * blockDim.x + threadIdx.x;
    if (i >= n) return;
    int g = batch[i];
    float4 v = ((const float4*)agg2)[i];
    unsafeAtomicAdd(&psum[g * 4 + 0], eluf(v.x + b2[0]));
    unsafeAtomicAdd(&psum[g * 4 + 1], eluf(v.y + b2[1]));
    unsafeAtomicAdd(&psum[g * 4 + 2], eluf(v.z + b2[2]));
    unsafeAtomicAdd(&psum[g * 4 + 3], eluf(v.w + b2[3]));
    unsafeAtomicAdd(&pcnt[g], 1.0f);
}

__global__ void k_final(const float* __restrict__ psum, const float* __restrict__ pcnt,
                        const float* __restrict__ W3, const float* __restrict__ b3,
                        float* __restrict__ out, int G) {
    int g = blockIdx.x * blockDim.x + threadIdx.x;
    if (g >= G) return;
    float c = fmaxf(pcnt[g], 1.0f);
    float acc = b3[0];
    acc += (psum[g * 4 + 0] / c) * W3[0];
    acc += (psum[g * 4 + 1] / c) * W3[1];
    acc += (psum[g * 4 + 2] / c) * W3[2];
    acc += (psum[g * 4 + 3] / c) * W3[3];
    out[g] = acc;
}

// ---------------- launch ----------------

static inline int cdiv(long a, long b) { return (int)((a + b - 1) / b); }
static inline size_t alignup(size_t v) { return (v + 255) & ~(size_t)255; }

extern "C" void kernel_launch(void* const* d_in, const int* in_sizes, int n_in,
                              void* d_out, int out_size, void* d_ws, size_t ws_size,
                              hipStream_t stream) {
    const float* x     = (const float*)d_in[0];
    const int*   ei    = (const int*)d_in[1];
    const int*   batch = (const int*)d_in[2];
    const float* W1    = (const float*)d_in[3];
    const float* b1    = (const float*)d_in[4];
    const float* W2    = (const float*)d_in[5];
    const float* b2    = (const float*)d_in[6];
    const float* W3    = (const float*)d_in[7];
    const float* b3    = (const float*)d_in[8];
    float* out = (float*)d_out;

    const int n = in_sizes[0] / 128;   // 100000 nodes
    const int e = in_sizes[1] / 2;     // 3.2M edges
    const int G = out_size;            // 256 graphs
    const int* row = ei;               // edge_index[0]
    const int* col = ei + e;           // edge_index[1]

    // workspace carve (~55 MB total)
    char* p = (char*)d_ws;
    float* dinv = (float*)p;  p += alignup((size_t)n * 4);
    float* t1   = (float*)p;  p += alignup((size_t)n * 64 * 4);
    float* agg1 = (float*)p;  p += alignup((size_t)n * 64 * 4);
    float* t2   = (float*)p;  p += alignup((size_t)n * 4 * 4);
    float* agg2 = (float*)p;  p += alignup((size_t)n * 4 * 4);
    float* psum = (float*)p;  p += alignup((size_t)G * 4 * 4);
    float* pcnt = (float*)p;  p += alignup((size_t)G * 4);
    int*   degi = (int*)p;    p += alignup((size_t)n * 4);
    (void)ws_size; (void)n_in;

    const int B = 256;

    // 1. zero counters
    k_zero<<<cdiv(n, B), B, 0, stream>>>(degi, psum, pcnt, n, G);
    // 2. degrees -> dinv (self-loop folded in)
    k_deg<<<cdiv(e, B), B, 0, stream>>>(col, degi, e);
    k_dinv<<<cdiv(n, B), B, 0, stream>>>(degi, dinv, n);
    // 3. t1 = x @ W1 (f32 WMMA)
    k_gemm1<<<cdiv(n, 128), B, 0, stream>>>(x, W1, t1, n);
    // 4. agg1 = t1 * dinv^2 (self-loop message == initializer)
    k_init64<<<cdiv((long)n * 16, B), B, 0, stream>>>(t1, dinv, agg1, n);
    // 5. edge scatter layer 1
    k_scatter1<<<cdiv((long)e * 16, B), B, 0, stream>>>(row, col, dinv, t1, agg1, e);
    // 6. h1 = elu(agg1 + b1), in place
    k_elu1<<<cdiv((long)n * 64, B), B, 0, stream>>>(agg1, b1, (long)n * 64);
    // 7. t2 = h1 @ W2 (f32 WMMA, N padded)
    k_gemm2<<<cdiv(n, 128), B, 0, stream>>>(agg1, W2, t2, n);
    // 8. agg2 = t2 * dinv^2, edge scatter layer 2
    k_init4<<<cdiv(n, B), B, 0, stream>>>(t2, dinv, agg2, n);
    k_scatter2<<<cdiv(e, B), B, 0, stream>>>(row, col, dinv, t2, agg2, e);
    // 9. pooling + final linear
    k_pool<<<cdiv(n, B), B, 0, stream>>>(agg2, b2, batch, psum, pcnt, n);
    k_final<<<cdiv(G, B), B, 0, stream>>>(psum, pcnt, W3, b3, out, G);
}